// ClauseInferModule_33646773797501
// MI455X (gfx1250) — compile-verified
//
#include <hip/hip_runtime.h>
#include <cstdint>

// Problem constants (reference: C=16, B=64, G=2048, S=8, L=4, gamma=1e-3, 3 steps)
#define C_   16
#define B_   64
#define G_   2048
#define S_   8
#define L_   4
#define TGG  32            // atoms (gg) per block tile in gather kernels
#define GAMMA 0.001f
#define INVG  1000.0f
#define NTOT  (C_ * G_ * B_)   // 2,097,152 elements of R

// ---------- monotonic float<->uint mapping for atomicMax on floats ----------
__device__ __forceinline__ unsigned f2key(float f) {
  unsigned u = __float_as_uint(f);
  return (u >> 31) ? ~u : (u | 0x80000000u);
}
__device__ __forceinline__ float key2f(unsigned k) {
  return __uint_as_float((k >> 31) ? (k & 0x7fffffffu) : ~k);
}

// ---------- CDNA5 async-to-LDS wait ----------
__device__ __forceinline__ void asyncWait0() {
#if __has_builtin(__builtin_amdgcn_s_wait_asynccnt)
  __builtin_amdgcn_s_wait_asynccnt(0);
#else
  asm volatile("s_wait_asynccnt 0" ::: "memory");
#endif
}

// Stage TGG*S_*L_ = 1024 indices (4 KB) into LDS: one b128 async copy per thread.
// VDST operand = LDS byte address (low 32 bits of generic LDS pointer, per ISA
// aperture mapping); VADDR = 64-bit global address.
__device__ __forceinline__ void stageIdx(const int* __restrict__ gsrc,
                                         int* sIdx, int tid) {
  unsigned ldsa = (unsigned)(uintptr_t)(sIdx + tid * 4);
  const int* ga = gsrc + tid * 4;
  asm volatile("global_load_async_to_lds_b128 %0, %1, off"
               :: "v"(ldsa), "v"(ga) : "memory");
  asyncWait0();
  __syncthreads();
}

// ---------- wave32 + block max reduction, then one atomic per block ----------
__device__ __forceinline__ void blockAtomicMax(float v, unsigned* dst) {
  #pragma unroll
  for (int off = 16; off > 0; off >>= 1)
    v = fmaxf(v, __shfl_xor(v, off, 32));
  __shared__ float smax[8];                     // 256 threads = 8 wave32
  int lane = threadIdx.x & 31, w = threadIdx.x >> 5;
  if (lane == 0) smax[w] = v;
  __syncthreads();
  if (threadIdx.x == 0) {
    float m = smax[0];
    #pragma unroll
    for (int i = 1; i < 8; ++i) m = fmaxf(m, smax[i]);
    atomicMax(dst, f2key(m));
  }
}

// ---------- init: R_t[c,g,b] = x[b,g]  (R stored transposed (C,G,B)) ----------
__global__ void __launch_bounds__(256) k_init(const float* __restrict__ x,
                                              float* __restrict__ Rt) {
  size_t i = ((size_t)blockIdx.x * 256 + threadIdx.x) * 4;   // 4 consecutive b
  int b = (int)(i & (B_ - 1));
  int g = (int)((i >> 6) % G_);
  float4 v;
  v.x = x[(b + 0) * G_ + g];
  v.y = x[(b + 1) * G_ + g];
  v.z = x[(b + 2) * G_ + g];
  v.w = x[(b + 3) * G_ + g];
  *(float4*)(Rt + i) = v;
}

__global__ void k_zero(unsigned* __restrict__ mx) {
  if (threadIdx.x < 2 * C_ + 1) mx[threadIdx.x] = 0u;
}

// ---------- stable soft-AND over L=4, two b-columns per lane (float2) ----------
__device__ __forceinline__ float2 softand4v2(const float* __restrict__ Rc,
                                             const int* __restrict__ q, int bb) {
  float2 a0 = *(const float2*)(Rc + q[0] * B_ + bb);
  float2 a1 = *(const float2*)(Rc + q[1] * B_ + bb);
  float2 a2 = *(const float2*)(Rc + q[2] * B_ + bb);
  float2 a3 = *(const float2*)(Rc + q[3] * B_ + bb);
  float mnx = fminf(fminf(a0.x, a1.x), fminf(a2.x, a3.x));
  float mny = fminf(fminf(a0.y, a1.y), fminf(a2.y, a3.y));
  float sx = __expf((mnx - a0.x) * INVG) + __expf((mnx - a1.x) * INVG)
           + __expf((mnx - a2.x) * INVG) + __expf((mnx - a3.x) * INVG);
  float sy = __expf((mny - a0.y) * INVG) + __expf((mny - a1.y) * INVG)
           + __expf((mny - a2.y) * INVG) + __expf((mny - a3.y) * INVG);
  float2 r;
  r.x = mnx - GAMMA * __logf(sx);              // -gamma*LSE(-x/gamma)
  r.y = mny - GAMMA * __logf(sy);
  return r;
}

// ---------- K1: per-clause max of soft-AND values (m1[c]) ----------
__global__ void __launch_bounds__(256) k_softand_max(
    const float* __restrict__ Rt, const int* __restrict__ I,
    unsigned* __restrict__ mx) {
  __shared__ int sIdx[TGG * S_ * L_];
  const int tid = threadIdx.x;
  const int c  = blockIdx.x / (G_ / TGG);
  const int g0 = (blockIdx.x % (G_ / TGG)) * TGG;
  stageIdx(I + ((size_t)c * G_ + g0) * (S_ * L_), sIdx, tid);

  const int lane = tid & 31, wv = tid >> 5;    // 8 waves; wave covers all B=64
  const int bb = lane * 2;
  const float* __restrict__ Rc = Rt + (size_t)c * G_ * B_;
  float lmax = -3.0e38f;
  for (int gl = wv; gl < TGG; gl += 8) {
    const int* ip = sIdx + gl * (S_ * L_);
    #pragma unroll
    for (int s = 0; s < S_; ++s) {
      float2 v = softand4v2(Rc, ip + s * L_, bb);
      lmax = fmaxf(lmax, fmaxf(v.x, v.y));
    }
  }
  blockAtomicMax(lmax, mx + c);
}

// ---------- K2: recompute soft-AND, normalize by m1, soft-OR over S -> Q; m2[c] ----------
__global__ void __launch_bounds__(256) k_softor_subst(
    const float* __restrict__ Rt, const int* __restrict__ I,
    unsigned* __restrict__ mx, float* __restrict__ Q) {
  __shared__ int sIdx[TGG * S_ * L_];
  const int tid = threadIdx.x;
  const int c  = blockIdx.x / (G_ / TGG);
  const int g0 = (blockIdx.x % (G_ / TGG)) * TGG;
  stageIdx(I + ((size_t)c * G_ + g0) * (S_ * L_), sIdx, tid);

  const float m1  = key2f(mx[c]);
  const float rn1 = (m1 > 1.0f) ? (1.0f / m1) : 1.0f;

  const int lane = tid & 31, wv = tid >> 5;
  const int bb = lane * 2;
  const float* __restrict__ Rc = Rt + (size_t)c * G_ * B_;
  float lmax = -3.0e38f;
  for (int gl = wv; gl < TGG; gl += 8) {
    const int* ip = sIdx + gl * (S_ * L_);
    float px[S_], py[S_];
    #pragma unroll
    for (int s = 0; s < S_; ++s) {
      float2 v = softand4v2(Rc, ip + s * L_, bb);
      px[s] = v.x * rn1;
      py[s] = v.y * rn1;
    }
    float mxx = px[0], mxy = py[0];
    #pragma unroll
    for (int s = 1; s < S_; ++s) { mxx = fmaxf(mxx, px[s]); mxy = fmaxf(mxy, py[s]); }
    float sumx = 0.0f, sumy = 0.0f;
    #pragma unroll
    for (int s = 0; s < S_; ++s) {
      sumx += __expf((px[s] - mxx) * INVG);
      sumy += __expf((py[s] - mxy) * INVG);
    }
    float2 s2;
    s2.x = mxx + GAMMA * __logf(sumx);         // gamma*LSE(p/gamma)
    s2.y = mxy + GAMMA * __logf(sumy);
    *(float2*)(Q + ((size_t)c * G_ + (g0 + gl)) * B_ + bb) = s2;
    lmax = fmaxf(lmax, fmaxf(s2.x, s2.y));
  }
  blockAtomicMax(lmax, mx + C_ + c);
}

// ---------- K3: merge R and r(R) pairwise soft-OR; global max m3 ----------
__global__ void __launch_bounds__(256) k_combine(
    const float* __restrict__ Rt, float* __restrict__ Q,
    unsigned* __restrict__ mx) {
  const size_t i = ((size_t)blockIdx.x * 256 + threadIdx.x) * 4;
  const int c = (int)(i / ((size_t)G_ * B_));
  const float m2  = key2f(mx[C_ + c]);
  const float rn2 = (m2 > 1.0f) ? (1.0f / m2) : 1.0f;
  float4 r = *(const float4*)(Rt + i);
  float4 q = *(const float4*)(Q + i);
  float o[4];
  float rr[4] = {r.x, r.y, r.z, r.w};
  float qq[4] = {q.x * rn2, q.y * rn2, q.z * rn2, q.w * rn2};
  float lmax = -3.0e38f;
  #pragma unroll
  for (int k = 0; k < 4; ++k) {
    float hi = fmaxf(rr[k], qq[k]), lo = fminf(rr[k], qq[k]);
    o[k] = hi + GAMMA * __logf(1.0f + __expf((lo - hi) * INVG));
    lmax = fmaxf(lmax, o[k]);
  }
  float4 ov = {o[0], o[1], o[2], o[3]};
  *(float4*)(Q + i) = ov;
  blockAtomicMax(lmax, mx + 2 * C_);
}

// ---------- K4: global renormalize -> new R ----------
__global__ void __launch_bounds__(256) k_norm(
    float* __restrict__ Rt, const float* __restrict__ Q,
    const unsigned* __restrict__ mx) {
  const size_t i = ((size_t)blockIdx.x * 256 + threadIdx.x) * 4;
  const float m3 = key2f(mx[2 * C_]);
  const float rn = (m3 > 1.0f) ? (1.0f / m3) : 1.0f;
  float4 q = *(const float4*)(Q + i);
  float4 v = {q.x * rn, q.y * rn, q.z * rn, q.w * rn};
  *(float4*)(Rt + i) = v;
}

// ---------- output: untranspose (C,G,B) -> (C,B,G) ----------
__global__ void __launch_bounds__(256) k_out(const float* __restrict__ Rt,
                                             float* __restrict__ out) {
  const size_t i = ((size_t)blockIdx.x * 256 + threadIdx.x) * 4;  // (c,b,g), g fastest
  int g = (int)(i % G_);                       // 4 consecutive g, same (c,b)
  int b = (int)((i / G_) & (B_ - 1));
  int c = (int)(i / ((size_t)G_ * B_));
  const float* base = Rt + ((size_t)c * G_ + g) * B_ + b;
  float4 v;
  v.x = base[0 * B_];
  v.y = base[1 * B_];
  v.z = base[2 * B_];
  v.w = base[3 * B_];
  *(float4*)(out + i) = v;
}

extern "C" void kernel_launch(void* const* d_in, const int* in_sizes, int n_in,
                              void* d_out, int out_size, void* d_ws, size_t ws_size,
                              hipStream_t stream) {
  (void)in_sizes; (void)n_in; (void)out_size; (void)ws_size;
  const float* x = (const float*)d_in[0];       // (B,G) f32
  const int*   I = (const int*)d_in[1];         // (C,G,S,L) i32
  (void)d_in[2];                                // infer_step: fixed 3 (static graph)

  float*    Rt = (float*)d_ws;                  // (C,G,B) transposed valuations
  float*    Q  = Rt + (size_t)NTOT;             // scratch / r(R)
  unsigned* mx = (unsigned*)(Q + (size_t)NTOT); // m1[16], m2[16], m3[1]

  k_init<<<NTOT / 1024, 256, 0, stream>>>(x, Rt);
  for (int step = 0; step < 3; ++step) {
    k_zero<<<1, 64, 0, stream>>>(mx);
    k_softand_max<<<C_ * (G_ / TGG), 256, 0, stream>>>(Rt, I, mx);
    k_softor_subst<<<C_ * (G_ / TGG), 256, 0, stream>>>(Rt, I, mx, Q);
    k_combine<<<NTOT / 1024, 256, 0, stream>>>(Rt, Q, mx);
    k_norm<<<NTOT / 1024, 256, 0, stream>>>(Rt, Q, mx);
  }
  k_out<<<NTOT / 1024, 256, 0, stream>>>(Rt, (float*)d_out);
}